// DetectionLoss_50603304682166
// MI455X (gfx1250) — compile-verified
//
#include <hip/hip_runtime.h>
#include <hip/hip_bf16.h>

#define FEPS 1e-6f

typedef float v2f __attribute__((ext_vector_type(2)));
typedef float v8f __attribute__((ext_vector_type(8)));

// ---------- exact 32-lane sum via V_WMMA_F32_16X16X4_F32 (B = ones) ----------
__device__ __forceinline__ float wave_sum_wmma(float x) {
#if __has_builtin(__builtin_amdgcn_wmma_f32_16x16x4_f32)
  v2f a; a[0] = x;   a[1] = 0.0f;   // A[m,k]: lane<16 -> k={0,1}, lane>=16 -> k={2,3}
  v2f b; b[0] = 1.0f; b[1] = 1.0f;  // B all ones -> D[m,n] = x_m + x_{m+16}
  v8f c = {};
  c = __builtin_amdgcn_wmma_f32_16x16x4_f32(false, a, false, b, (short)0, c,
                                            false, false);
  float s = c[0] + c[1] + c[2] + c[3] + c[4] + c[5] + c[6] + c[7];
  s += __shfl_xor(s, 16, 32);       // half-sums live in lane groups [0,16)/[16,32)
  return s;
#else
  for (int off = 16; off > 0; off >>= 1) x += __shfl_xor(x, off, 32);
  return x;
#endif
}

// block reduction; result valid on threadIdx.x == 0
__device__ __forceinline__ float block_sum(float x, float* red, int nwaves) {
  int lane = threadIdx.x & 31, w = threadIdx.x >> 5;
  float s = wave_sum_wmma(x);
  __syncthreads();                 // protect red[] reuse across calls
  if (lane == 0) red[w] = s;
  __syncthreads();
  float tot = 0.0f;
  if (threadIdx.x == 0)
    for (int i = 0; i < nwaves; i++) tot += red[i];
  return tot;
}

__device__ __forceinline__ unsigned long long wave_max_u64(unsigned long long v) {
  for (int off = 16; off > 0; off >>= 1) {
    unsigned long long o = __shfl_xor(v, off, 32);
    v = (o > v) ? o : v;
  }
  return v;
}

__device__ __forceinline__ float box_iou(float4 a, float4 b) {
  float x1 = fmaxf(a.x, b.x), y1 = fmaxf(a.y, b.y);
  float x2 = fminf(a.z, b.z), y2 = fminf(a.w, b.w);
  float inter = fmaxf(x2 - x1, 0.0f) * fmaxf(y2 - y1, 0.0f);
  float a1 = (a.z - a.x) * (a.w - a.y);
  float a2 = (b.z - b.x) * (b.w - b.y);
  return inter / (a1 + a2 - inter + FEPS);
}

__device__ __forceinline__ float giou_plus_l1(float4 p, float4 t) {
  float x1 = fmaxf(p.x, t.x), y1 = fmaxf(p.y, t.y);
  float x2 = fminf(p.z, t.z), y2 = fminf(p.w, t.w);
  float inter = fmaxf(x2 - x1, 0.0f) * fmaxf(y2 - y1, 0.0f);
  float a1 = (p.z - p.x) * (p.w - p.y);
  float a2 = (t.z - t.x) * (t.w - t.y);
  float uni = a1 + a2 - inter;
  float iou = inter / (uni + FEPS);
  float ex1 = fminf(p.x, t.x), ey1 = fminf(p.y, t.y);
  float ex2 = fmaxf(p.z, t.z), ey2 = fmaxf(p.w, t.w);
  float enc = (ex2 - ex1) * (ey2 - ey1);
  float giou = iou - (enc - uni) / (enc + FEPS);
  float l1 = fabsf(p.x - t.x) + fabsf(p.y - t.y) + fabsf(p.z - t.z) + fabsf(p.w - t.w);
  return (1.0f - giou) + 0.5f * 0.25f * l1;
}

__device__ __forceinline__ float focal_fn(float conf, bool pos) {
  float pt = pos ? conf : (1.0f - conf);
  float af = pos ? 0.25f : 0.75f;
  float om = 1.0f - pt;                                  // unclipped, like reference
  float ptc = fminf(fmaxf(pt, FEPS), 1.0f - FEPS);
  return -af * om * om * logf(ptc);
}

// ---------------- init: zero per-call state ----------------
__global__ void k_init(unsigned* forcedU, int nF4, int* cand, int nCand,
                       int* accI, int nAcc) {
  int i = blockIdx.x * blockDim.x + threadIdx.x;
  if (i < nF4) forcedU[i] = 0u;
  if (i < nCand) cand[i] = 0;
  if (i < nAcc) accI[i] = 0;
}

// ---- k1: per-block top-3 per target (u64 keys) + candidate counts ----
__global__ void k_topk_partial(const float* __restrict__ anchors,
                               const float* __restrict__ tboxes,
                               unsigned long long* __restrict__ part, // [B,nblk,T,3]
                               int* __restrict__ cand,                // [B,T]
                               int A, int T, int nblk) {
  int img = blockIdx.y;
  __shared__ float4 tg[64];
  __shared__ int candS[64];
  __shared__ unsigned long long wtop[32 * 3];
  int tid = threadIdx.x;
  if (tid < T) {
    const float* tp = tboxes + ((size_t)img * T + tid) * 4;
    tg[tid] = make_float4(tp[0], tp[1], tp[2], tp[3]);
    candS[tid] = 0;
  }
  __syncthreads();

  int a = blockIdx.x * blockDim.x + tid;
  bool valid = (a < A);
  float4 ab = make_float4(0, 0, 0, 0);
  if (valid) {
    const float* ap = anchors + (size_t)a * 4;
    ab = make_float4(ap[0], ap[1], ap[2], ap[3]);
  }
  int lane = tid & 31, w = tid >> 5;
  int nw = blockDim.x >> 5;

  for (int t = 0; t < T; t++) {
    float iou = valid ? box_iou(ab, tg[t]) : -1.0f;
    unsigned long long bal = __ballot(valid && (iou > 0.3f));
    if (lane == 0) atomicAdd(&candS[t], (int)__popcll(bal));

    unsigned long long key = 0ull;
    if (valid)
      key = ((unsigned long long)__float_as_uint(iou) << 32) |
            (unsigned long long)(0xFFFFFFFFu - (unsigned)a);  // lower idx wins ties

    unsigned long long k = key;                 // wave-level top-3 (sorted desc)
    for (int r = 0; r < 3; r++) {
      unsigned long long m = wave_max_u64(k);
      if (lane == 0) wtop[w * 3 + r] = m;
      if (k == m) k = 0ull;
    }
    __syncthreads();
    if (w == 0) {                               // merge 32 sorted-3 lists
      unsigned long long L0 = 0, L1 = 0, L2 = 0;
      if (lane < nw) { L0 = wtop[lane*3]; L1 = wtop[lane*3+1]; L2 = wtop[lane*3+2]; }
      int ptr = 0;
      unsigned long long out0 = 0, out1 = 0, out2 = 0;
      for (int r = 0; r < 3; r++) {
        unsigned long long head = (ptr == 0) ? L0 : (ptr == 1) ? L1 : (ptr == 2) ? L2 : 0ull;
        unsigned long long m = wave_max_u64(head);
        if (ptr < 3 && head == m) ptr++;
        if (r == 0) out0 = m; else if (r == 1) out1 = m; else out2 = m;
      }
      if (lane == 0) {
        size_t base = (((size_t)img * nblk + blockIdx.x) * T + t) * 3;
        part[base] = out0; part[base + 1] = out1; part[base + 2] = out2;
      }
    }
    __syncthreads();
  }
  if (tid < T) atomicAdd(&cand[img * T + tid], candS[tid]);
}

// ---- k2: final per-target top-3, set forced flags (top k_t, k_t=max(1,min(3,cand))) ----
__global__ void k_topk_final(const unsigned long long* __restrict__ part,
                             const int* __restrict__ cand,
                             unsigned char* __restrict__ forced,
                             int A, int T, int nblk) {
  int img = blockIdx.x;
  int lane = threadIdx.x & 31, w = threadIdx.x >> 5;
  if (w >= T) return;
  int t = w;
  const unsigned long long* p = part + (size_t)img * nblk * T * 3;
  int P = nblk * 3;
  unsigned long long L0 = 0, L1 = 0, L2 = 0;
  for (int i = lane; i < P; i += 32) {
    int b = i / 3, r = i % 3;
    unsigned long long k = p[((size_t)b * T + t) * 3 + r];
    if (k > L0) { L2 = L1; L1 = L0; L0 = k; }
    else if (k > L1) { L2 = L1; L1 = k; }
    else if (k > L2) { L2 = k; }
  }
  int ptr = 0;
  unsigned long long out[3];
  for (int r = 0; r < 3; r++) {
    unsigned long long head = (ptr == 0) ? L0 : (ptr == 1) ? L1 : (ptr == 2) ? L2 : 0ull;
    unsigned long long m = wave_max_u64(head);
    if (ptr < 3 && head == m) ptr++;
    out[r] = m;
  }
  if (lane == 0) {
    int c = cand[img * T + t];
    int kk = c < 1 ? 1 : (c > 3 ? 3 : c);
    for (int r = 0; r < kk; r++) {
      unsigned idx = 0xFFFFFFFFu - (unsigned)(out[r] & 0xFFFFFFFFull);
      if (idx < (unsigned)A) forced[(size_t)img * A + idx] = 1;
    }
  }
}

// ---- k3: classify anchors, focal, bbox loss, WMMA block reductions ----
__global__ void k_classify(const float* __restrict__ bbox, const float* __restrict__ conf,
                           const float* __restrict__ anchors, const float* __restrict__ tboxes,
                           const unsigned char* __restrict__ forced,
                           unsigned* __restrict__ focalbits,      // [B,A] (0 => not-negative)
                           float* __restrict__ blockPart,         // [B,nblk3,2]
                           int* __restrict__ accI,                // [B,2] {num_pos,num_neg}
                           int A, int T, int nblk3) {
  int img = blockIdx.y;
  __shared__ float4 tg[64];
  __shared__ float red[8];
  int tid = threadIdx.x;
  if (tid < T) {
    const float* tp = tboxes + ((size_t)img * T + tid) * 4;
    tg[tid] = make_float4(tp[0], tp[1], tp[2], tp[3]);
  }
  __syncthreads();

  int a = blockIdx.x * blockDim.x + tid;
  bool valid = (a < A);
  float maxiou = -1.0f; int bt = 0;
  if (valid) {
    const float* ap = anchors + (size_t)a * 4;
    float4 ab = make_float4(ap[0], ap[1], ap[2], ap[3]);
    for (int t = 0; t < T; t++) {
      float iou = box_iou(ab, tg[t]);
      if (iou > maxiou) { maxiou = iou; bt = t; }   // first-max like argmax
    }
  }
  bool pos = false, neg = false;
  float f = 0.0f, bl = 0.0f;
  if (valid) {
    pos = (maxiou >= 0.5f) || (forced[(size_t)img * A + a] != 0);
    neg = (maxiou < 0.4f) && !pos;
    float cf = conf[(size_t)img * A + a];
    f = focal_fn(cf, pos);
    focalbits[(size_t)img * A + a] = neg ? __float_as_uint(f) : 0u;
    if (pos) {
      const float* bp = bbox + ((size_t)img * A + (size_t)a) * 4;
      float4 pb = make_float4(bp[0], bp[1], bp[2], bp[3]);
      bl = giou_plus_l1(pb, tg[bt]);
    }
  }
  float sp = block_sum(pos ? f : 0.0f, red, 8);
  float sb = block_sum(bl, red, 8);
  float cp = block_sum(pos ? 1.0f : 0.0f, red, 8);
  float cn = block_sum(neg ? 1.0f : 0.0f, red, 8);
  if (tid == 0) {
    size_t bo = ((size_t)img * nblk3 + blockIdx.x) * 2;
    blockPart[bo + 0] = sp;
    blockPart[bo + 1] = sb;
    atomicAdd(&accI[img * 2 + 0], (int)(cp + 0.5f));
    atomicAdd(&accI[img * 2 + 1], (int)(cn + 0.5f));
  }
}

// ---- k4: per-image exact top-K negative selection (radix select) + finalize ----
__global__ void k_select(const unsigned* __restrict__ focalbits,
                         const float* __restrict__ blockPart,
                         const int* __restrict__ accI,
                         float* __restrict__ imgLoss,
                         int A, int nblk3) {
  int img = blockIdx.x;
  const unsigned* fb = focalbits + (size_t)img * A;
  __shared__ int hist[256];
  __shared__ int sh_sel, sh_krem;
  __shared__ float red[32];
  int tid = threadIdx.x, nthr = blockDim.x;

  int np = accI[img * 2 + 0];
  int nn = accI[img * 2 + 1];
  int npc = np < 1 ? 1 : np;
  int ratio = A / npc; if (ratio > 3) ratio = 3;
  int maxneg = ratio * np;

  unsigned vbits = 0u; int krem = 0; int cnt_sel;
  if (nn > maxneg) {                    // uniform branch per block
    cnt_sel = maxneg;
    int k = maxneg;                     // 1-indexed rank of threshold value
    unsigned prefix = 0u;
    for (int pass = 0; pass < 4; pass++) {
      int shift = 24 - pass * 8;
      unsigned mask_hi = (pass == 0) ? 0u : (0xFFFFFFFFu << (shift + 8));
      for (int i = tid; i < 256; i += nthr) hist[i] = 0;
      __syncthreads();
      for (int i = tid; i < A; i += nthr) {
        unsigned v = fb[i];
        if (v != 0u && (v & mask_hi) == prefix)
          atomicAdd(&hist[(v >> shift) & 0xFF], 1);
      }
      __syncthreads();
      if (tid == 0) {
        int c = 0, sel = 0;
        for (int d = 255; d >= 0; d--) {
          int h = hist[d];
          if (c + h >= k) { sel = d; break; }
          c += h;
        }
        sh_sel = sel; sh_krem = k - c;
      }
      __syncthreads();
      prefix |= ((unsigned)sh_sel) << shift;
      k = sh_krem;
    }
    vbits = prefix; krem = k;           // krem copies of the threshold value
  } else {
    cnt_sel = nn; vbits = 0u; krem = 0; // keep all negatives
  }
  __syncthreads();

  // sum of focal over strictly-greater-than-threshold negatives
  float s = 0.0f;
  for (int i = tid; i < A; i += nthr) {
    unsigned v = fb[i];
    if (v > vbits) s += __uint_as_float(v);
  }
  float sum_gt = block_sum(s, red, 32);

  // combine per-block partials (deterministic order)
  float xp = 0.0f, xb = 0.0f;
  for (int i = tid; i < nblk3; i += nthr) {
    size_t bo = ((size_t)img * nblk3 + i) * 2;
    xp += blockPart[bo + 0];
    xb += blockPart[bo + 1];
  }
  float sum_pos = block_sum(xp, red, 32);
  float sum_bbox = block_sum(xb, red, 32);

  if (tid == 0) {
    float sum_sel = sum_gt + (float)krem * __uint_as_float(vbits);
    int denom = np + cnt_sel; if (denom < 1) denom = 1;
    float conf_loss = (sum_pos + sum_sel) / (float)denom;
    float bbox_loss = sum_bbox / (float)npc;
    imgLoss[img] = conf_loss + bbox_loss;
  }
}

__global__ void k_final(const float* __restrict__ imgLoss, float* __restrict__ out, int B) {
  if (blockIdx.x == 0 && threadIdx.x == 0) {
    float s = 0.0f;
    for (int i = 0; i < B; i++) s += imgLoss[i];
    out[0] = s / (float)B;
  }
}

extern "C" void kernel_launch(void* const* d_in, const int* in_sizes, int n_in,
                              void* d_out, int out_size, void* d_ws, size_t ws_size,
                              hipStream_t stream) {
  const float* bbox    = (const float*)d_in[0];   // [B,A,4]
  const float* conf    = (const float*)d_in[1];   // [B,A]
  const float* anchors = (const float*)d_in[2];   // [A,4]
  const float* tboxes  = (const float*)d_in[3];   // [B,T,4]

  int A = in_sizes[2] / 4;
  int B = in_sizes[1] / A;
  int T = in_sizes[3] / (4 * B);

  int nblk1 = (A + 1023) / 1024;   // k1 blocks per image
  int nblk3 = (A + 255) / 256;     // k3 blocks per image

  // --- workspace carve (16B aligned sections) ---
  char* w = (char*)d_ws;
  unsigned* focalbits = (unsigned*)w;            w += (size_t)B * A * 4;
  unsigned long long* part = (unsigned long long*)w;
                                                 w += (size_t)B * nblk1 * T * 3 * 8;
  unsigned char* forced = (unsigned char*)w;     w += (((size_t)B * A + 15) / 16) * 16;
  int* cand = (int*)w;                           w += (size_t)B * T * 4;
  int* accI = (int*)w;                           w += (size_t)B * 2 * 4 + 8;
  float* blockPart = (float*)w;                  w += (size_t)B * nblk3 * 2 * 4;
  float* imgLoss = (float*)w;                    w += (size_t)B * 4;
  float* out = (float*)d_out;

  int nF4 = (B * A) / 4;
  int initN = nF4 > B * T ? nF4 : B * T;
  k_init<<<(initN + 255) / 256, 256, 0, stream>>>((unsigned*)forced, nF4,
                                                  cand, B * T, accI, B * 2);
  k_topk_partial<<<dim3(nblk1, B), 1024, 0, stream>>>(anchors, tboxes, part, cand,
                                                      A, T, nblk1);
  k_topk_final<<<B, T * 32, 0, stream>>>(part, cand, forced, A, T, nblk1);
  k_classify<<<dim3(nblk3, B), 256, 0, stream>>>(bbox, conf, anchors, tboxes, forced,
                                                 focalbits, blockPart, accI, A, T, nblk3);
  k_select<<<B, 1024, 0, stream>>>(focalbits, blockPart, accI, imgLoss, A, nblk3);
  k_final<<<1, 32, 0, stream>>>(imgLoss, out, B);
}